// LearnableMultiheadSelfAttention_21500606284071
// MI455X (gfx1250) — compile-verified
//
#include <hip/hip_runtime.h>
#include <hip/hip_bf16.h>
#include <stdint.h>

typedef float v2f __attribute__((ext_vector_type(2)));
typedef float v8f __attribute__((ext_vector_type(8)));

#define S_LEN   256
#define BATCH   4
#define DMODEL  1024
#define NHEAD   16
#define DHEAD   64
#define KSEL    4
#define TLEN    2304              // (mem_num+1)*x_len
#define SCALE   0.125f            // 1/sqrt(64)

__device__ __forceinline__ v8f wmma_f32(v2f a, v2f b, v8f c) {
  // V_WMMA_F32_16X16X4_F32 : D(16x16 f32) = A(16x4 f32) * B(4x16 f32) + C
  return __builtin_amdgcn_wmma_f32_16x16x4_f32(false, a, false, b, (short)0, c, false, false);
}

// ---------------------------------------------------------------------------
// f32 WMMA GEMM, 32x32 output tile per wave (2x2 register blocking):
//   out[m,o] = sum_d Arow(m)[d] * W[o,d]
//   Arow(m) = (m < split) ? A1 + m*D : A2 + (m-split)*D   (fused concat gather)
// Per K-step (K=4): 2 A pairs + 2 B pairs loaded, 4 WMMAs -> 1.0 loads/WMMA.
// A lane layout (16x4 f32 A): lane (h,l): A[m0+l, k + 2h + {0,1}]
// B lane layout (4x16 f32 B): lane (h,l): B[k + 2h + {0,1}, o0+l] = W[o0+l, ...]
// D layout: lane (h,l), vgpr r -> out[row0 + r + 8h, col0 + l]
// ---------------------------------------------------------------------------
__global__ void gemm_wmma_f32(const float* __restrict__ A1, const float* __restrict__ A2,
                              int split, const float* __restrict__ W,
                              float* __restrict__ out, int M, int O, int D) {
  int wave = blockIdx.x * (blockDim.x >> 5) + (threadIdx.x >> 5);
  int oT = O >> 5;                     // 32-wide o tiles
  int nT = (M >> 5) * oT;
  if (wave >= nT) return;
  int mt = wave / oT, ot = wave - mt * oT;
  int lane = threadIdx.x & 31;
  int hh = lane >> 4, l = lane & 15;
  int m0 = mt << 5, o0 = ot << 5;

  int ar0 = m0 + l, ar1 = m0 + 16 + l;
  const float* ap0 = ((ar0 < split) ? (A1 + (size_t)ar0 * D)
                                    : (A2 + (size_t)(ar0 - split) * D)) + 2 * hh;
  const float* ap1 = ((ar1 < split) ? (A1 + (size_t)ar1 * D)
                                    : (A2 + (size_t)(ar1 - split) * D)) + 2 * hh;
  const float* wp0 = W + (size_t)(o0 + l) * D + 2 * hh;
  const float* wp1 = W + (size_t)(o0 + 16 + l) * D + 2 * hh;

  v8f acc00 = {0.f,0.f,0.f,0.f,0.f,0.f,0.f,0.f};
  v8f acc01 = acc00, acc10 = acc00, acc11 = acc00;
#pragma unroll 2
  for (int k = 0; k < D; k += 4) {
    v2f a0 = *(const v2f*)(ap0 + k);
    v2f a1 = *(const v2f*)(ap1 + k);
    v2f b0 = *(const v2f*)(wp0 + k);
    v2f b1 = *(const v2f*)(wp1 + k);
    acc00 = wmma_f32(a0, b0, acc00);
    acc01 = wmma_f32(a0, b1, acc01);
    acc10 = wmma_f32(a1, b0, acc10);
    acc11 = wmma_f32(a1, b1, acc11);
  }
  float* op00 = out + (size_t)(m0 + (hh << 3)) * O + o0 + l;
  float* op01 = op00 + 16;
  float* op10 = out + (size_t)(m0 + 16 + (hh << 3)) * O + o0 + l;
  float* op11 = op10 + 16;
#pragma unroll
  for (int r = 0; r < 8; ++r) {
    op00[(size_t)r * O] = acc00[r];
    op01[(size_t)r * O] = acc01[r];
    op10[(size_t)r * O] = acc10[r];
    op11[(size_t)r * O] = acc11[r];
  }
}

// ---------------------------------------------------------------------------
// BD_raw[b][n][i][t] = sum_d (q[i,b,n,d] + pos_bias_v[n,d]) * r[t,b,n,d]
// One wave per (b, n, i-tile16, t-tile64): A operand built once per K-step and
// reused over 4 t-subtiles (4 WMMAs per K-step, 1.25 loads/WMMA).
// ---------------------------------------------------------------------------
__global__ void bd_wmma(const float* __restrict__ qbuf, const float* __restrict__ rbuf,
                        const float* __restrict__ vbias, float* __restrict__ bd) {
  const int TT = TLEN / 64;  // 36 t-tiles of 64
  int wave = blockIdx.x * (blockDim.x >> 5) + (threadIdx.x >> 5);
  if (wave >= BATCH * NHEAD * 16 * TT) return;
  int tt = wave % TT; int tmp = wave / TT;
  int it = tmp & 15; tmp >>= 4;
  int n = tmp & 15;  int b = tmp >> 4;
  int lane = threadIdx.x & 31;
  int hh = lane >> 4, l = lane & 15;
  int t0 = tt * 64;

  const float* qrow = qbuf + ((size_t)((it * 16 + l) * BATCH + b)) * DMODEL + n * DHEAD + 2 * hh;
  const float* bias = vbias + n * DHEAD + 2 * hh;
  const float* rr0 = rbuf + ((size_t)((t0      + l) * BATCH + b)) * DMODEL + n * DHEAD + 2 * hh;
  const float* rr1 = rbuf + ((size_t)((t0 + 16 + l) * BATCH + b)) * DMODEL + n * DHEAD + 2 * hh;
  const float* rr2 = rbuf + ((size_t)((t0 + 32 + l) * BATCH + b)) * DMODEL + n * DHEAD + 2 * hh;
  const float* rr3 = rbuf + ((size_t)((t0 + 48 + l) * BATCH + b)) * DMODEL + n * DHEAD + 2 * hh;

  v8f acc0 = {0.f,0.f,0.f,0.f,0.f,0.f,0.f,0.f};
  v8f acc1 = acc0, acc2 = acc0, acc3 = acc0;
#pragma unroll
  for (int kk = 0; kk < DHEAD; kk += 4) {
    v2f a;
    a.x = qrow[kk] + bias[kk];
    a.y = qrow[kk + 1] + bias[kk + 1];
    v2f b0 = *(const v2f*)(rr0 + kk);
    v2f b1 = *(const v2f*)(rr1 + kk);
    v2f b2 = *(const v2f*)(rr2 + kk);
    v2f b3 = *(const v2f*)(rr3 + kk);
    acc0 = wmma_f32(a, b0, acc0);
    acc1 = wmma_f32(a, b1, acc1);
    acc2 = wmma_f32(a, b2, acc2);
    acc3 = wmma_f32(a, b3, acc3);
  }
  float* op = bd + ((size_t)((b * NHEAD + n) * S_LEN + it * 16 + (hh << 3))) * TLEN + t0 + l;
#pragma unroll
  for (int r = 0; r < 8; ++r) {
    op[(size_t)r * TLEN]      = acc0[r];
    op[(size_t)r * TLEN + 16] = acc1[r];
    op[(size_t)r * TLEN + 32] = acc2[r];
    op[(size_t)r * TLEN + 48] = acc3[r];
  }
}

// ---------------------------------------------------------------------------
// Score + softmax-over-K + attn_matrix + diagonal probs.
// One wave per (b, n, i-tile, j-tile). A operand (q+u) preloaded to registers
// and reused for all K=4 selected blocks; BD gathered per-element with the
// exact rel-shift index; softmax over K entirely in registers.
// ---------------------------------------------------------------------------
__global__ void score_softmax(const float* __restrict__ qbuf, const float* __restrict__ kvbuf,
                              const float* __restrict__ ubias, const float* __restrict__ bd,
                              const int* __restrict__ indices,
                              float* __restrict__ pdiag, float* __restrict__ amat) {
  int wave = blockIdx.x * (blockDim.x >> 5) + (threadIdx.x >> 5);
  if (wave >= BATCH * NHEAD * 16 * 16) return;
  int jt = wave & 15; int tmp = wave >> 4;
  int it = tmp & 15;  tmp >>= 4;
  int n = tmp & 15;   int b = tmp >> 4;
  int lane = threadIdx.x & 31;
  int hh = lane >> 4, l = lane & 15;
  int i0 = it << 4, j0 = jt << 4;

  const float* qrow = qbuf + ((size_t)((i0 + l) * BATCH + b)) * DMODEL + n * DHEAD + 2 * hh;
  const float* ub = ubias + n * DHEAD + 2 * hh;
  v2f aq[16];
#pragma unroll
  for (int kk = 0; kk < 16; ++kk) {
    aq[kk].x = qrow[kk * 4] + ub[kk * 4];
    aq[kk].y = qrow[kk * 4 + 1] + ub[kk * 4 + 1];
  }

  const float* bdbase = bd + (size_t)((b * NHEAD + n) * S_LEN) * TLEN;

  float sc[KSEL][8];
#pragma unroll
  for (int ks = 0; ks < KSEL; ++ks) {
    int blk = indices[ks * BATCH + b];
    const float* krow = kvbuf + ((size_t)((blk * S_LEN + j0 + l) * BATCH + b)) * (2 * DMODEL)
                        + n * DHEAD + 2 * hh;
    v8f acc = {0.f,0.f,0.f,0.f,0.f,0.f,0.f,0.f};
#pragma unroll
    for (int kk = 0; kk < 16; ++kk) {
      v2f bm = *(const v2f*)(krow + kk * 4);
      acc = wmma_f32(aq[kk], bm, acc);
    }
#pragma unroll
    for (int r = 0; r < 8; ++r) {
      int i = i0 + r + (hh << 3);
      int jj = blk * S_LEN + j0 + l;
      unsigned idx = 256u + (unsigned)i * 2304u + (unsigned)jj;  // rel-shift index
      unsigned ip = idx / 2305u;
      unsigned jp = idx - ip * 2305u;
      float bdv = (jp == 0u) ? 0.f : bdbase[(size_t)ip * TLEN + (jp - 1u)];
      sc[ks][r] = (acc[r] + bdv) * SCALE;
    }
  }

  // softmax over the K selected blocks, per element, all in registers
  float p[KSEL][8];
#pragma unroll
  for (int r = 0; r < 8; ++r) {
    float m = sc[0][r];
#pragma unroll
    for (int ks = 1; ks < KSEL; ++ks) m = fmaxf(m, sc[ks][r]);
    float s = 0.f;
#pragma unroll
    for (int ks = 0; ks < KSEL; ++ks) { float e = expf(sc[ks][r] - m); p[ks][r] = e; s += e; }
    float inv = 1.f / s;
#pragma unroll
    for (int ks = 0; ks < KSEL; ++ks) p[ks][r] *= inv;
  }

  // attn_matrix[i, ks, n] += sum_j p / 1024  (mean over j and b)
#pragma unroll
  for (int ks = 0; ks < KSEL; ++ks) {
#pragma unroll
    for (int r = 0; r < 8; ++r) {
      float s = p[ks][r];
      s += __shfl_xor(s, 1);
      s += __shfl_xor(s, 2);
      s += __shfl_xor(s, 4);
      s += __shfl_xor(s, 8);   // sum over the 16 j-columns within each lane-half
      if (l == 0) {
        int i = i0 + r + (hh << 3);
        atomicAdd(amat + i * 64 + ks * 16 + n, s * (1.0f / 1024.0f));
      }
    }
  }

  // diagonal probs p[i, ks, i, b, n] for the attn_vec path
  if (it == jt) {
#pragma unroll
    for (int ks = 0; ks < KSEL; ++ks) {
#pragma unroll
      for (int r = 0; r < 8; ++r) {
        if (l == r + (hh << 3)) {  // row == col in this tile
          pdiag[(size_t)(i0 + l) * 256 + ks * 64 + b * 16 + n] = p[ks][r];
        }
      }
    }
  }
}

// ---------------------------------------------------------------------------
// attn_vec[i,b,n,d] = sum_k pdiag[i,k,b,n] * v[idx(k,b), i, b, n, d]
// ---------------------------------------------------------------------------
__global__ void attn_vec_kernel(const float* __restrict__ pd, const float* __restrict__ kvb,
                                const int* __restrict__ idxs, float* __restrict__ av) {
  int gid = blockIdx.x * blockDim.x + threadIdx.x;
  if (gid >= S_LEN * BATCH * DMODEL) return;
  int row = gid >> 10, col = gid & 1023;
  int i = row >> 2, b = row & 3;
  int n = col >> 6;
  float acc = 0.f;
#pragma unroll
  for (int ks = 0; ks < KSEL; ++ks) {
    int blk = idxs[ks * BATCH + b];
    float pv = pd[(size_t)i * 256 + ks * 64 + b * 16 + n];
    acc += pv * kvb[((size_t)((blk * S_LEN + i) * BATCH + b)) * (2 * DMODEL) + DMODEL + col];
  }
  av[gid] = acc;
}

// ---------------------------------------------------------------------------
// y = x + attn_out ; LayerNorm over D=1024 (biased variance), one block/row.
// ---------------------------------------------------------------------------
__global__ void ln_kernel(const float* __restrict__ x, const float* __restrict__ aout,
                          const float* __restrict__ gamma, const float* __restrict__ beta,
                          float* __restrict__ out) {
  int row = blockIdx.x;        // 0..1023 = s*4 + b
  int tid = threadIdx.x;       // 256 threads
  __shared__ float ls[256], ls2[256];
  float y[4];
  float s = 0.f, s2 = 0.f;
#pragma unroll
  for (int k = 0; k < 4; ++k) {
    int c = tid + k * 256;
    float v = x[(size_t)row * DMODEL + c] + aout[(size_t)row * DMODEL + c];
    y[k] = v; s += v; s2 += v * v;
  }
  ls[tid] = s; ls2[tid] = s2;
  __syncthreads();
  for (int o = 128; o > 0; o >>= 1) {
    if (tid < o) { ls[tid] += ls[tid + o]; ls2[tid] += ls2[tid + o]; }
    __syncthreads();
  }
  float mean = ls[0] * (1.0f / 1024.0f);
  float var = ls2[0] * (1.0f / 1024.0f) - mean * mean;
  float inv = rsqrtf(var + 1e-5f);
#pragma unroll
  for (int k = 0; k < 4; ++k) {
    int c = tid + k * 256;
    out[(size_t)row * DMODEL + c] = (y[k] - mean) * inv * gamma[c] + beta[c];
  }
}

__global__ void zero_kernel(float* p, int n) {
  int i = blockIdx.x * blockDim.x + threadIdx.x;
  if (i < n) p[i] = 0.f;
}

// ---------------------------------------------------------------------------
extern "C" void kernel_launch(void* const* d_in, const int* in_sizes, int n_in,
                              void* d_out, int out_size, void* d_ws, size_t ws_size,
                              hipStream_t stream) {
  const float* x    = (const float*)d_in[0];   // [256,4,1024]
  const float* pos  = (const float*)d_in[1];   // [2304,4,1024]
  const float* mem  = (const float*)d_in[2];   // [8,256,4,1024] == rows 0..8191 of c
  const int*   idxs = (const int*)d_in[3];     // [4,4]
  const float* ub   = (const float*)d_in[4];   // [16,64]
  const float* vb   = (const float*)d_in[5];   // [16,64]
  const float* Wq   = (const float*)d_in[6];   // [1024,1024]
  const float* Wkv  = (const float*)d_in[7];   // [2048,1024]
  const float* Wr   = (const float*)d_in[8];   // [1024,1024]
  const float* Wo   = (const float*)d_in[9];   // [1024,1024]
  const float* gam  = (const float*)d_in[10];
  const float* bet  = (const float*)d_in[11];
  float* out = (float*)d_out;

  float* ws  = (float*)d_ws;
  float* qb  = ws;                  // 1,048,576  : q  [1024,1024]
  float* rb  = qb + 1048576;        // 9,437,184  : r  [9216,1024]
  float* kvb = rb + 9437184;        // 18,874,368 : kv [9216,2048]
  float* bdb = kvb + 18874368;      // 37,748,736 : BD [4][16][256][2304]
  float* pd  = bdb + 37748736;      // 65,536     : pdiag [256][4][4][16]
  float* av  = pd + 65536;          // 1,048,576  : attn_vec [1024,1024]
  float* ao  = av + 1048576;        // 1,048,576  : attn_out [1024,1024]

  float* amat = out + 1048576;      // attn_matrix [256,4,16] appended after output

  const int BIG = 1 << 30;

  zero_kernel<<<64, 256, 0, stream>>>(amat, 16384);

  // q = x @ Wq^T
  { int tiles = (1024 / 32) * (1024 / 32);
    gemm_wmma_f32<<<(tiles + 7) / 8, 256, 0, stream>>>(x, x, BIG, Wq, qb, 1024, 1024, DMODEL); }
  // r = pos_emb @ Wr^T
  { int tiles = (9216 / 32) * (1024 / 32);
    gemm_wmma_f32<<<(tiles + 7) / 8, 256, 0, stream>>>(pos, pos, BIG, Wr, rb, 9216, 1024, DMODEL); }
  // kv = concat(memory, x) @ Wkv^T   (split row 8192)
  { int tiles = (9216 / 32) * (2048 / 32);
    gemm_wmma_f32<<<(tiles + 7) / 8, 256, 0, stream>>>(mem, x, 8192, Wkv, kvb, 9216, 2048, DMODEL); }

  // BD_raw = (q + v_bias) . r^T  per (b, n)
  { int waves = BATCH * NHEAD * 16 * (TLEN / 64);  // 36864
    bd_wmma<<<(waves + 7) / 8, 256, 0, stream>>>(qb, rb, vb, bdb); }

  // scores (AC via WMMA + shifted BD), softmax over K, attn_matrix, diag probs
  { int waves = BATCH * NHEAD * 16 * 16;           // 16384
    score_softmax<<<(waves + 7) / 8, 256, 0, stream>>>(qb, kvb, ub, bdb, idxs, pd, amat); }

  // attn_vec gather
  attn_vec_kernel<<<(S_LEN * BATCH * DMODEL + 255) / 256, 256, 0, stream>>>(pd, kvb, idxs, av);

  // attn_out = attn_vec @ Wo^T
  { int tiles = (1024 / 32) * (1024 / 32);
    gemm_wmma_f32<<<(tiles + 7) / 8, 256, 0, stream>>>(av, av, BIG, Wo, ao, 1024, 1024, DMODEL); }

  // residual + LayerNorm
  ln_kernel<<<1024, 256, 0, stream>>>(x, ao, gam, bet, out);
}